// SpatialTransformer_2DTo2D_real_64106681860170
// MI455X (gfx1250) — compile-verified
//
#include <hip/hip_runtime.h>

// Bilinear feature sampling (SpatialTransformer 2D->2D) for gfx1250.
// Memory-bound gather: wave-per-point, async global->LDS double-buffered
// staging of the 4 corner rows, NT streaming stores for the output.

typedef float v4f __attribute__((ext_vector_type(4)));

constexpr int FH    = 90;
constexpr int FW    = 160;
constexpr int FDIM  = 512;
constexpr int NV    = 19200;           // 120 * 160 ground-plane points
constexpr int CHUNK = 128;             // floats per wave-chunk (32 lanes x float4)
constexpr int NCHUNK = FDIM / CHUNK;   // 4
constexpr int WAVES_PER_BLOCK = 8;     // 256 threads

// Issue one async 16B-per-lane copy: global -> LDS (GV mode, saddr = off).
// VDST operand is the per-lane LDS byte address in a VGPR.
__device__ __forceinline__ void async_ld_b128(const float* gaddr, const float* lds_ptr) {
    unsigned lds_addr = (unsigned)(unsigned long long)lds_ptr;  // low 32 bits = LDS byte offset
    asm volatile("global_load_async_to_lds_b128 %0, %1, off"
                 :: "v"(lds_addr), "v"(gaddr)
                 : "memory");
}

__global__ __launch_bounds__(256) void st2d_bilinear_kernel(
    const float* __restrict__ feat,   // [nR, FH, FW, FDIM]
    const float* __restrict__ imx,    // [nR, NV]
    const float* __restrict__ imy,    // [nR, NV]
    float* __restrict__ out,          // [nR, NV, FDIM]
    int nPoints) {

    // Per-wave double buffer: 2 x 4 corners x 128 floats = 4KB -> 32KB/block.
    __shared__ __align__(16) float lbuf[WAVES_PER_BLOCK][2][4][CHUNK];

    const int tid  = threadIdx.x;
    const int lane = tid & 31;
    const int wave = tid >> 5;
    const int p    = blockIdx.x * WAVES_PER_BLOCK + wave;  // flat (r, v) point index
    if (p >= nPoints) return;                              // wave-uniform guard

    const int r = p / NV;

    // ---- wave-uniform bilinear setup (matches reference exactly) ----
    float ix = imx[p];
    float iy = imy[p];
    ix = fminf(fmaxf(ix, 0.0f), (float)(FW - 1));
    iy = fminf(fmaxf(iy, 0.0f), (float)(FH - 1));
    const float x0f = floorf(ix), y0f = floorf(iy);
    const int   x0  = (int)x0f,   y0  = (int)y0f;
    const int   x1  = min(x0 + 1, FW - 1);
    const int   y1  = min(y0 + 1, FH - 1);
    const float x1f = (float)x1,  y1f = (float)y1;
    const float wx1 = x1f - ix, wx0 = ix - x0f;
    const float wy1 = y1f - iy, wy0 = iy - y0f;
    const float wa = wx1 * wy1;   // (y0, x0)
    const float wb = wx1 * wy0;   // (y1, x0)
    const float wc = wx0 * wy1;   // (y0, x1)
    const float wd = wx0 * wy0;   // (y1, x1)

    const float* base = feat + (size_t)r * FH * FW * FDIM;
    const float* rows[4];
    rows[0] = base + ((size_t)y0 * FW + x0) * FDIM;
    rows[1] = base + ((size_t)y1 * FW + x0) * FDIM;
    rows[2] = base + ((size_t)y0 * FW + x1) * FDIM;
    rows[3] = base + ((size_t)y1 * FW + x1) * FDIM;

    float* optr = out + (size_t)p * FDIM;
    const int loff = lane * 4;   // this lane's float4 slot within a chunk

    // ---- prologue: async-stage chunk 0 into buffer 0 (4 x 512B coalesced) ----
    #pragma unroll
    for (int k = 0; k < 4; ++k)
        async_ld_b128(rows[k] + loff, &lbuf[wave][0][k][loff]);

    // ---- pipelined chunk loop: prefetch c+1 while blending c ----
    #pragma unroll
    for (int c = 0; c < NCHUNK; ++c) {
        const int buf = c & 1;
        if (c + 1 < NCHUNK) {
            const int noff = (c + 1) * CHUNK + loff;
            #pragma unroll
            for (int k = 0; k < 4; ++k)
                async_ld_b128(rows[k] + noff, &lbuf[wave][buf ^ 1][k][loff]);
            // 8 async ops outstanding; <=4 left => chunk c is complete (in-order).
            asm volatile("s_wait_asynccnt 4" ::: "memory");
        } else {
            asm volatile("s_wait_asynccnt 0" ::: "memory");
        }

        // Blend from LDS (ds_load_b128, lane-contiguous -> conflict-free).
        const v4f A = *(const v4f*)&lbuf[wave][buf][0][loff];
        const v4f B = *(const v4f*)&lbuf[wave][buf][1][loff];
        const v4f C = *(const v4f*)&lbuf[wave][buf][2][loff];
        const v4f D = *(const v4f*)&lbuf[wave][buf][3][loff];
        const v4f O = wa * A + wb * B + wc * C + wd * D;

        // Output is streamed exactly once: non-temporal store keeps the
        // feature tensor resident in the 192MB L2.
        __builtin_nontemporal_store(O, (v4f*)(optr + c * CHUNK + loff));
    }
}

extern "C" void kernel_launch(void* const* d_in, const int* in_sizes, int n_in,
                              void* d_out, int out_size, void* d_ws, size_t ws_size,
                              hipStream_t stream) {
    const float* feat = (const float*)d_in[0];  // [4, 90, 160, 512] f32
    const float* imx  = (const float*)d_in[1];  // [4, 19200] f32
    const float* imy  = (const float*)d_in[2];  // [4, 19200] f32
    float* out        = (float*)d_out;          // [4, 19200, 512] f32

    const int nPoints = in_sizes[1];            // nR * nV = 76800
    const int blocks  = (nPoints + WAVES_PER_BLOCK - 1) / WAVES_PER_BLOCK;

    st2d_bilinear_kernel<<<blocks, WAVES_PER_BLOCK * 32, 0, stream>>>(
        feat, imx, imy, out, nPoints);
}